// SparseConvCausalAttention_33243046871697
// MI455X (gfx1250) — compile-verified
//
#include <hip/hip_runtime.h>
#include <float.h>

// ---------------------------------------------------------------------------
// SparseConvCausalAttention on MI455X (gfx1250), fp32 via V_WMMA_F32_16X16X4_F32
// b=2, n=1152, dim=512, H=8, DH=64, text=128 (tokens 1024..1151 as q/k/v_text),
// img=1024 (tokens 128..1151), KS=5, SCALE=1/8 folded into q at qkv projection.
//
// v2: B staged TRANSPOSED in LDS so every WMMA B-fragment is one 8B-aligned
// ds_load_b64 (no v_dual_mov repacking); 128x128 GEMM tiles (8 acc/wave, 8x
// A-frag reuse); contiguous global fragments loaded as float2.
// ---------------------------------------------------------------------------

typedef float v2f __attribute__((ext_vector_type(2)));
typedef float v8f __attribute__((ext_vector_type(8)));

__device__ __forceinline__ v8f wmma_f32(v2f a, v2f b, v8f c) {
  // D = A(16x4) * B(4x16) + C(16x16), fp32, wave32
  return __builtin_amdgcn_wmma_f32_16x16x4_f32(false, a, false, b, (short)0, c,
                                               false, false);
}

#define NTOK 1152
#define DIM 512
#define QKVW 1536
#define NROWS 2304 /* 2*1152 */
#define NEGF (-3.402823466e38f)

// ---------------------------------------------------------------------------
// Kernel A: qkv = X(2304x512) @ W(512x1536); q columns (<512) scaled by 1/8.
// Block: 256 thr = 8 waves; tile 128 rows x 128 cols; wave w -> rows 16w..16w+15.
// A rows padded to 34 floats (8B-aligned pairs, conflict-free 34-stride banks);
// B stored transposed [n][k] so B-frags are contiguous b64 loads.
// ---------------------------------------------------------------------------
__global__ void __launch_bounds__(256) qkv_gemm(const float* __restrict__ X,
                                                const float* __restrict__ W,
                                                float* __restrict__ Y) {
  __shared__ float As[128][34];
  __shared__ float BsT[128][34];
  const int rowBase = blockIdx.x * 128;
  const int colBase = blockIdx.y * 128;
  const int t = threadIdx.x;
  const int wave = t >> 5, lane = t & 31;
  const int m16 = lane & 15, kg = lane >> 4;

  v8f acc[8] = {};
  for (int k0 = 0; k0 < 512; k0 += 32) {
    for (int i = t; i < 128 * 32; i += 256) {
      int r = i >> 5, c = i & 31;  // consecutive t -> consecutive c: coalesced
      As[r][c] = X[(rowBase + r) * 512 + k0 + c];
    }
    for (int i = t; i < 32 * 128; i += 256) {
      int c = i >> 7, n = i & 127;  // consecutive t -> consecutive n: coalesced
      BsT[n][c] = W[(k0 + c) * QKVW + colBase + n];
    }
    __syncthreads();
#pragma unroll
    for (int kk = 0; kk < 32; kk += 4) {
      v2f a = *(const v2f*)&As[wave * 16 + m16][kk + 2 * kg];
#pragma unroll
      for (int t8 = 0; t8 < 8; ++t8) {
        v2f bb = *(const v2f*)&BsT[t8 * 16 + m16][kk + 2 * kg];
        acc[t8] = wmma_f32(a, bb, acc[t8]);
      }
    }
    __syncthreads();
  }
  const float scale = (colBase < 512) ? 0.125f : 1.0f;
#pragma unroll
  for (int t8 = 0; t8 < 8; ++t8)
#pragma unroll
    for (int r = 0; r < 8; ++r) {
      int row = rowBase + wave * 16 + r + 8 * kg;
      int col = colBase + t8 * 16 + m16;
      Y[row * QKVW + col] = acc[t8][r] * scale;
    }
}

// ---------------------------------------------------------------------------
// Kernel B: text self-attention. Tokens 1024..1151 attend to each other.
// grid = 32 blocks: bh = bx>>1 (16), half = bx&1 (64 queries each).
// Output goes to final tokens 0..127.
// ---------------------------------------------------------------------------
__global__ void __launch_bounds__(256) text_attn(const float* __restrict__ QKV,
                                                 float* __restrict__ AO) {
  __shared__ float dots[64][132];
  const int bh = blockIdx.x >> 1;
  const int qh = blockIdx.x & 1;
  const int b = bh >> 3, h = bh & 7;
  const int qrowbase = qh * 64;
  const int t = threadIdx.x;
  const int wave = t >> 5, lane = t & 31;
  const int m16 = lane & 15, kg = lane >> 4;

  const int txt0 = (b * NTOK + 1024) * QKVW;  // first text token row in qkv
  const float* qtext = QKV + txt0 + h * 64;
  const float* ktext = QKV + txt0 + 512 + h * 64;
  const float* vtext = QKV + txt0 + 1024 + h * 64;

  // Phase 1: dots = q_text @ k_text^T  (64x128 per block)
  {
    const int rt = wave & 3, ch = wave >> 2;
    v8f acc[4] = {};
    const float* qr = qtext + (qrowbase + rt * 16 + m16) * QKVW;
#pragma unroll
    for (int kk = 0; kk < 64; kk += 4) {
      v2f a = *(const v2f*)(qr + kk + 2 * kg);
#pragma unroll
      for (int t4 = 0; t4 < 4; ++t4) {
        int key = ch * 64 + t4 * 16 + m16;
        v2f bb = *(const v2f*)(ktext + key * QKVW + kk + 2 * kg);
        acc[t4] = wmma_f32(a, bb, acc[t4]);
      }
    }
#pragma unroll
    for (int t4 = 0; t4 < 4; ++t4)
#pragma unroll
      for (int r = 0; r < 8; ++r)
        dots[rt * 16 + r + 8 * kg][ch * 64 + t4 * 16 + m16] = acc[t4][r];
  }
  __syncthreads();

  // Phase 2: softmax over 128 keys
  if (t < 64) {
    float mx = NEGF;
    for (int j = 0; j < 128; ++j) mx = fmaxf(mx, dots[t][j]);
    float s = 0.0f;
    for (int j = 0; j < 128; ++j) {
      float e = expf(dots[t][j] - mx);
      dots[t][j] = e;
      s += e;
    }
    float inv = 1.0f / s;
    for (int j = 0; j < 128; ++j) dots[t][j] *= inv;
  }
  __syncthreads();

  // Phase 3: out = attn @ v_text (64x64), waves 0..3
  if (wave < 4) {
    v8f acc[4] = {};
#pragma unroll 4
    for (int kk = 0; kk < 128; kk += 4) {
      v2f a = *(const v2f*)&dots[wave * 16 + m16][kk + 2 * kg];
#pragma unroll
      for (int t4 = 0; t4 < 4; ++t4) {
        v2f bb;
        bb.x = vtext[(kk + 2 * kg) * QKVW + t4 * 16 + m16];
        bb.y = vtext[(kk + 2 * kg + 1) * QKVW + t4 * 16 + m16];
        acc[t4] = wmma_f32(a, bb, acc[t4]);
      }
    }
#pragma unroll
    for (int t4 = 0; t4 < 4; ++t4)
#pragma unroll
      for (int r = 0; r < 8; ++r) {
        int tok = qrowbase + wave * 16 + r + 8 * kg;  // final tokens 0..127
        AO[(b * NTOK + tok) * DIM + h * 64 + t4 * 16 + m16] = acc[t4][r];
      }
  }
}

// ---------------------------------------------------------------------------
// Kernel C: image attention. Queries = tokens 128..1151 (image index 0..1023),
// keys = 128 text tokens (WMMA) + 25 local 5x5 neighbors (VALU, masked).
// grid = 256 blocks: bh = bx>>4 (16), qblk = bx&15 (64 queries each).
// Output goes to final tokens 128..1151.
// ---------------------------------------------------------------------------
__global__ void __launch_bounds__(256) img_attn(const float* __restrict__ QKV,
                                                float* __restrict__ AO) {
  __shared__ float dots[64][156];    // 128 text + 25 local
  __shared__ float outloc[64][68];   // local-neighbor contribution
  const int bh = blockIdx.x >> 4;
  const int qblk = blockIdx.x & 15;
  const int b = bh >> 3, h = bh & 7;
  const int qbase = qblk * 64;
  const int t = threadIdx.x;
  const int wave = t >> 5, lane = t & 31;
  const int m16 = lane & 15, kg = lane >> 4;

  const int img0 = (b * NTOK + 128) * QKVW;
  const int txt0 = (b * NTOK + 1024) * QKVW;
  const float* qimg = QKV + img0 + h * 64;
  const float* kimg = QKV + img0 + 512 + h * 64;
  const float* vimg = QKV + img0 + 1024 + h * 64;
  const float* ktext = QKV + txt0 + 512 + h * 64;
  const float* vtext = QKV + txt0 + 1024 + h * 64;

  // Phase 1a: dots[:, 0:128] = q_img @ k_text^T via WMMA (8 waves)
  {
    const int rt = wave & 3, ch = wave >> 2;
    v8f acc[4] = {};
    const float* qr = qimg + (qbase + rt * 16 + m16) * QKVW;
#pragma unroll
    for (int kk = 0; kk < 64; kk += 4) {
      v2f a = *(const v2f*)(qr + kk + 2 * kg);
#pragma unroll
      for (int t4 = 0; t4 < 4; ++t4) {
        int key = ch * 64 + t4 * 16 + m16;
        v2f bb = *(const v2f*)(ktext + key * QKVW + kk + 2 * kg);
        acc[t4] = wmma_f32(a, bb, acc[t4]);
      }
    }
#pragma unroll
    for (int t4 = 0; t4 < 4; ++t4)
#pragma unroll
      for (int r = 0; r < 8; ++r)
        dots[rt * 16 + r + 8 * kg][ch * 64 + t4 * 16 + m16] = acc[t4][r];
  }

  // Phase 1b: dots[:, 128:153] = masked local logits (VALU gather)
  for (int e = t; e < 64 * 25; e += 256) {
    int row = e / 25, j = e % 25;
    int qi = qbase + row;
    int y = qi >> 5, x = qi & 31;
    int ny = y + j / 5 - 2, nx = x + (j % 5) - 2;
    float val = 0.0f;  // zero-padded neighbor: kidx==img_len passes mask, dot==0
    if (ny >= 0 && ny < 32 && nx >= 0 && nx < 32) {
      int kidx = ny * 32 + nx;
      if (qi <= kidx) {
        const float* qr = qimg + qi * QKVW;
        const float* kr = kimg + kidx * QKVW;
        float s = 0.0f;
#pragma unroll 8
        for (int d = 0; d < 64; ++d) s += qr[d] * kr[d];
        val = s;
      } else {
        val = NEGF;
      }
    }
    dots[row][128 + j] = val;
  }
  __syncthreads();

  // Phase 2: softmax over 153 logits per row
  if (t < 64) {
    float mx = NEGF;
    for (int j = 0; j < 153; ++j) mx = fmaxf(mx, dots[t][j]);
    float s = 0.0f;
    for (int j = 0; j < 153; ++j) {
      float e = expf(dots[t][j] - mx);
      dots[t][j] = e;
      s += e;
    }
    float inv = 1.0f / s;
    for (int j = 0; j < 153; ++j) dots[t][j] *= inv;
  }
  __syncthreads();

  // Phase 3: waves 0-3 -> WMMA attn_txt @ v_text; waves 4-7 -> local v gather
  v8f acc[4] = {};
  if (wave < 4) {
#pragma unroll 4
    for (int kk = 0; kk < 128; kk += 4) {
      v2f a = *(const v2f*)&dots[wave * 16 + m16][kk + 2 * kg];
#pragma unroll
      for (int t4 = 0; t4 < 4; ++t4) {
        v2f bb;
        bb.x = vtext[(kk + 2 * kg) * QKVW + t4 * 16 + m16];
        bb.y = vtext[(kk + 2 * kg + 1) * QKVW + t4 * 16 + m16];
        acc[t4] = wmma_f32(a, bb, acc[t4]);
      }
    }
  } else {
    int t2 = t - 128;
    for (int e = t2; e < 64 * 64; e += 128) {
      int row = e >> 6, col = e & 63;
      int qi = qbase + row;
      int y = qi >> 5, x = qi & 31;
      float s = 0.0f;
#pragma unroll
      for (int j = 0; j < 25; ++j) {
        int ny = y + j / 5 - 2, nx = x + (j % 5) - 2;
        if (ny >= 0 && ny < 32 && nx >= 0 && nx < 32) {
          int kidx = ny * 32 + nx;
          s += dots[row][128 + j] * vimg[kidx * QKVW + col];
        }
      }
      outloc[row][col] = s;
    }
  }
  __syncthreads();

  if (wave < 4) {
#pragma unroll
    for (int t4 = 0; t4 < 4; ++t4)
#pragma unroll
      for (int r = 0; r < 8; ++r) {
        int lrow = wave * 16 + r + 8 * kg;
        int tok = 128 + qbase + lrow;  // final tokens 128..1151
        AO[(b * NTOK + tok) * DIM + h * 64 + t4 * 16 + m16] =
            acc[t4][r] + outloc[lrow][t4 * 16 + m16];
      }
  }
}

// ---------------------------------------------------------------------------
// Kernel D: Y = AO(2304x512) @ Wout(512x512) + bias; 128x128 tiles, BsT layout.
// ---------------------------------------------------------------------------
__global__ void __launch_bounds__(256) out_proj(const float* __restrict__ AO,
                                                const float* __restrict__ W,
                                                const float* __restrict__ bias,
                                                float* __restrict__ Y) {
  __shared__ float As[128][34];
  __shared__ float BsT[128][34];
  const int rowBase = blockIdx.x * 128;
  const int colBase = blockIdx.y * 128;
  const int t = threadIdx.x;
  const int wave = t >> 5, lane = t & 31;
  const int m16 = lane & 15, kg = lane >> 4;

  v8f acc[8] = {};
  for (int k0 = 0; k0 < 512; k0 += 32) {
    for (int i = t; i < 128 * 32; i += 256) {
      int r = i >> 5, c = i & 31;
      As[r][c] = AO[(rowBase + r) * DIM + k0 + c];
    }
    for (int i = t; i < 32 * 128; i += 256) {
      int c = i >> 7, n = i & 127;
      BsT[n][c] = W[(k0 + c) * DIM + colBase + n];
    }
    __syncthreads();
#pragma unroll
    for (int kk = 0; kk < 32; kk += 4) {
      v2f a = *(const v2f*)&As[wave * 16 + m16][kk + 2 * kg];
#pragma unroll
      for (int t8 = 0; t8 < 8; ++t8) {
        v2f bb = *(const v2f*)&BsT[t8 * 16 + m16][kk + 2 * kg];
        acc[t8] = wmma_f32(a, bb, acc[t8]);
      }
    }
    __syncthreads();
  }
#pragma unroll
  for (int t8 = 0; t8 < 8; ++t8)
#pragma unroll
    for (int r = 0; r < 8; ++r) {
      int row = rowBase + wave * 16 + r + 8 * kg;
      int col = colBase + t8 * 16 + m16;
      Y[row * DIM + col] = acc[t8][r] + bias[col];
    }
}

// ---------------------------------------------------------------------------
extern "C" void kernel_launch(void* const* d_in, const int* in_sizes, int n_in,
                              void* d_out, int out_size, void* d_ws,
                              size_t ws_size, hipStream_t stream) {
  const float* x = (const float*)d_in[0];      // (2,1152,512)
  const float* w_qkv = (const float*)d_in[1];  // (512,1536)
  const float* w_out = (const float*)d_in[2];  // (512,512)
  const float* b_out = (const float*)d_in[3];  // (512,)
  float* out = (float*)d_out;                  // (2,1152,512)

  float* qkv = (float*)d_ws;              // 2304*1536 floats (q pre-scaled)
  float* ao = qkv + (size_t)NROWS * QKVW; // 2304*512 floats (attention output)

  qkv_gemm<<<dim3(18, 12), 256, 0, stream>>>(x, w_qkv, qkv);
  text_attn<<<32, 256, 0, stream>>>(qkv, ao);
  img_attn<<<256, 256, 0, stream>>>(qkv, ao);
  out_proj<<<dim3(18, 4), 256, 0, stream>>>(ao, w_out, b_out, out);
}